// ScaledDotProductAttention_11562051960985
// MI455X (gfx1250) — compile-verified
//
#include <hip/hip_runtime.h>

typedef __attribute__((ext_vector_type(2))) float v2f;
typedef __attribute__((ext_vector_type(4))) float v4f;
typedef __attribute__((ext_vector_type(8))) float v8f;

#define ATTN_B 4
#define ATTN_H 16
#define ATTN_S 2048
#define ATTN_D 64
#define NKT (ATTN_S / 16)      // 128 k-tiles of 16 columns
#define QSCALE 0.125f          // 1/sqrt(64)
#define NEG_BIG (-1.0e9f)

// D = A(16x4) * B(4x16) + C(16x16), fp32.  8-arg form:
// (neg_a, A, neg_b, B, c_mod, C, reuse_a, reuse_b)
__device__ __forceinline__ v8f wmma_f32(v2f a, v2f b, v8f c) {
  return __builtin_amdgcn_wmma_f32_16x16x4_f32(false, a, false, b, (short)0, c,
                                               false, false);
}

__global__ __launch_bounds__(256) void sdpa_fused_kernel(
    const float* __restrict__ Q, const float* __restrict__ K,
    const float* __restrict__ V, const int* __restrict__ Mask,
    float* __restrict__ Out, float* __restrict__ Attn) {
  // Per-wave 16x16 P staging tile, COLUMN-major [col][row]:
  // store path = 2x ds_store_b128 per lane, readback conflict-free.
  __shared__ float lds_p[8][16][16];

  const int lane  = threadIdx.x & 31;
  const int wave  = threadIdx.x >> 5;
  const int qblk  = blockIdx.x & 15;   // 16 q-blocks of 128 rows per (b,h)
  const int bh    = blockIdx.x >> 4;   // 0..63
  const int b     = bh >> 4;           // H == 16
  const int qbase = qblk * 128 + wave * 16;

  const float* q   = Q + (size_t)bh * ATTN_S * ATTN_D;
  const float* k   = K + (size_t)bh * ATTN_S * ATTN_D;
  const float* v   = V + (size_t)bh * ATTN_S * ATTN_D;
  const int*   msk = Mask + (size_t)b * ATTN_S * ATTN_S;
  float* attn = Attn + (size_t)bh * ATTN_S * ATTN_S;
  float* out  = Out + (size_t)bh * ATTN_S * ATTN_D;

  // Fragment lane geometry (32-bit WMMA layouts):
  //  A 16x4:  lane -> (m = lane&15, k = 2*(lane>>4) + j) for VGPR j
  //  B 4x16:  lane -> (k = 2*(lane>>4) + j, n = lane&15)
  //  C 16x16: VGPR r, lane -> (m = r + 8*(lane>>4), n = lane&15)
  const int mrow  = lane & 15;
  const int ncol  = lane & 15;
  const int khalf = (lane >> 4);       // 0 or 1
  const int kb    = khalf << 1;        // 0 or 2
  const int rhalf = khalf << 3;        // 0 or 8

  // ---- Q tile as 16 chained K=4 A-fragments (lives in VGPRs all kernel) ----
  v2f aq[16];
  {
    const v2f* qrow = (const v2f*)(q + (size_t)(qbase + mrow) * ATTN_D);
    #pragma unroll
    for (int c = 0; c < 16; ++c) aq[c] = qrow[c * 2 + khalf];
  }

  // Lane-local online softmax state: each lane only ever sees columns
  // (kt*16 + ncol) of a row, so no cross-lane traffic until the epilogue.
  float lmax[8], lsum[8];
  #pragma unroll
  for (int r = 0; r < 8; ++r) {
    lmax[r] = -3.0e38f;
    lsum[r] = 0.0f;
  }

  // =============== Pass A: streaming lane-local statistics =================
  for (int kt = 0; kt < NKT; ++kt) {
    const float* kcol = k + (size_t)(kt * 16 + ncol) * ATTN_D;
    __builtin_prefetch(kcol + 16 * ATTN_D, 0, 0);  // next K tile
    const v2f* kcol2 = (const v2f*)kcol;

    v8f acc0 = {}, acc1 = {};   // 2-way ILP in the XDL chain
    #pragma unroll
    for (int c = 0; c < 8; ++c) {
      acc0 = wmma_f32(aq[2 * c + 0], kcol2[(2 * c + 0) * 2 + khalf], acc0);
      acc1 = wmma_f32(aq[2 * c + 1], kcol2[(2 * c + 1) * 2 + khalf], acc1);
    }

    const int colbase = kt * 16;
    #pragma unroll
    for (int r = 0; r < 8; ++r) {
      const int qr = qbase + rhalf + r;
      float s = (acc0[r] + acc1[r]) * QSCALE;
      if (msk[(size_t)qr * ATTN_S + colbase + ncol] == 0) s = NEG_BIG;
      const float mn = fmaxf(lmax[r], s);
      lsum[r] = lsum[r] * __expf(lmax[r] - mn) + __expf(s - mn);
      lmax[r] = mn;
    }
  }

  // ---- one-shot cross-lane combine: 8 rows x (4+4) shfl total ----
  float rmax[8], rinv[8];
  #pragma unroll
  for (int r = 0; r < 8; ++r) {
    float m = lmax[r];
    #pragma unroll
    for (int off = 1; off < 16; off <<= 1)
      m = fmaxf(m, __shfl_xor(m, off, 32));
    float sum = lsum[r] * __expf(lmax[r] - m);
    #pragma unroll
    for (int off = 1; off < 16; off <<= 1) sum += __shfl_xor(sum, off, 32);
    rmax[r] = m;
    rinv[r] = 1.0f / sum;
  }

  // ============ Pass B: recompute, emit attn, accumulate O = P*V ===========
  v8f oacc[4];
  #pragma unroll
  for (int d = 0; d < 4; ++d) oacc[d] = (v8f){};

  float* lp = &lds_p[wave][0][0];   // [col][row], stride 16

  for (int kt = 0; kt < NKT; ++kt) {
    const float* kcol = k + (size_t)(kt * 16 + ncol) * ATTN_D;
    __builtin_prefetch(kcol + 16 * ATTN_D, 0, 0);
    const v2f* kcol2 = (const v2f*)kcol;

    v8f acc0 = {}, acc1 = {};
    #pragma unroll
    for (int c = 0; c < 8; ++c) {
      acc0 = wmma_f32(aq[2 * c + 0], kcol2[(2 * c + 0) * 2 + khalf], acc0);
      acc1 = wmma_f32(aq[2 * c + 1], kcol2[(2 * c + 1) * 2 + khalf], acc1);
    }

    const int colbase = kt * 16;
    v4f pv0, pv1;
    #pragma unroll
    for (int r = 0; r < 8; ++r) {
      const int qr = qbase + rhalf + r;
      float s = (acc0[r] + acc1[r]) * QSCALE;
      if (msk[(size_t)qr * ATTN_S + colbase + ncol] == 0) s = NEG_BIG;
      const float p = __expf(s - rmax[r]) * rinv[r];
      attn[(size_t)qr * ATTN_S + colbase + ncol] = p;   // coalesced b32
      if (r < 4) pv0[r] = p; else pv1[r - 4] = p;
    }
    // Stage P column-major: lane owns column ncol, rows rhalf..rhalf+7
    // are contiguous -> two ds_store_b128. Same-wave LDS is in-order
    // (DScnt) and each wave owns a private tile: no barrier needed.
    *(v4f*)(lp + ncol * 16 + rhalf + 0) = pv0;
    *(v4f*)(lp + ncol * 16 + rhalf + 4) = pv1;

    // O(16x64) += P(16x16) * V(16x64): 4 K-chunks x 4 N-tiles of WMMA.
    const float* vbase = v + (size_t)colbase * ATTN_D;
    #pragma unroll
    for (int c = 0; c < 4; ++c) {
      v2f pa;   // A-frag of P: (m = mrow, k = c*4 + kb + j)
      pa.x = lp[(c * 4 + kb + 0) * 16 + mrow];
      pa.y = lp[(c * 4 + kb + 1) * 16 + mrow];
      const float* vr0 = vbase + (size_t)(c * 4 + kb + 0) * ATTN_D;
      const float* vr1 = vbase + (size_t)(c * 4 + kb + 1) * ATTN_D;
      #pragma unroll
      for (int d = 0; d < 4; ++d) {
        v2f vb;
        vb.x = vr0[d * 16 + ncol];
        vb.y = vr1[d * 16 + ncol];
        oacc[d] = wmma_f32(pa, vb, oacc[d]);
      }
    }
  }

  // ---- write O tile ----
  #pragma unroll
  for (int d = 0; d < 4; ++d) {
    #pragma unroll
    for (int r = 0; r < 8; ++r)
      out[(size_t)(qbase + rhalf + r) * ATTN_D + d * 16 + ncol] = oacc[d][r];
  }
}

extern "C" void kernel_launch(void* const* d_in, const int* in_sizes, int n_in,
                              void* d_out, int out_size, void* d_ws,
                              size_t ws_size, hipStream_t stream) {
  (void)in_sizes; (void)n_in; (void)out_size; (void)d_ws; (void)ws_size;
  const float* q    = (const float*)d_in[0];
  const float* k    = (const float*)d_in[1];
  const float* v    = (const float*)d_in[2];
  const int*   mask = (const int*)d_in[3];

  float* out  = (float*)d_out;                                    // [B,H,S,D]
  float* attn = out + (size_t)ATTN_B * ATTN_H * ATTN_S * ATTN_D;  // [B,H,S,S]

  // grid = B*H * (S/128) blocks; 8 wave32 per block, 16 q-rows per wave.
  dim3 grid(ATTN_B * ATTN_H * (ATTN_S / 128));
  sdpa_fused_kernel<<<grid, 256, 0, stream>>>(q, k, v, mask, out, attn);
}